// NWHead_65377992180609
// MI455X (gfx1250) — compile-verified
//
#include <hip/hip_runtime.h>
#include <hip/hip_bf16.h>
#include <math.h>

typedef __attribute__((ext_vector_type(2))) float v2f;
typedef __attribute__((ext_vector_type(8))) float v8f;

#define NB 64     // batches
#define NS 8192   // supports
#define ND 256    // embed dim
#define NC 100    // classes
#define JT 8      // support tiles per batch
#define TS (NS / JT)            // 1024 supports per tile
#define K1_THREADS 256
#define K1_WAVES (K1_THREADS / 32)
#define SUPW 16                  // supports per wave per WMMA pass
#define PASSES (TS / (K1_WAVES * SUPW))   // 1024 / 128 = 8

// workspace layout (floats):
//   [0, NB*NS)                     : w_j = exp(-dist) per (batch, support)
//   [WS_SUMS, +NB*JT)              : per-tile partial sums of w
//   [WS_POUT, +NB*JT*128)          : per-tile partial outputs (128-padded classes)
static constexpr size_t WS_SUMS = (size_t)NB * NS;
static constexpr size_t WS_POUT = WS_SUMS + (size_t)NB * JT;

// ---------------------------------------------------------------------------
// Kernel 1: weights w = exp(-||q - s||) via WMMA dot products.
// One block = (batch, support-tile of 1024). 8 waves, each wave does 16
// supports per pass with V_WMMA_F32_16X16X4_F32, 8 passes.
// ---------------------------------------------------------------------------
__global__ __launch_bounds__(K1_THREADS)
void nw_dist_kernel(const float* __restrict__ qf,
                    const float* __restrict__ sup,
                    float* __restrict__ ws) {
    __shared__ float qlds[ND];
    __shared__ float red[K1_THREADS];
    __shared__ float wpart[2 * K1_WAVES];

    const int tid   = threadIdx.x;
    const int jt    = blockIdx.x;
    const int batch = blockIdx.y;

    // stage q into LDS and compute ||q||^2 (deterministic tree reduction)
    const float qv = qf[batch * ND + tid];
    qlds[tid] = qv;
    red[tid]  = qv * qv;
    __syncthreads();
    for (int off = K1_THREADS / 2; off > 0; off >>= 1) {
        if (tid < off) red[tid] += red[tid + off];
        __syncthreads();
    }
    const float qn = red[0];

    const int lane = tid & 31;
    const int wave = tid >> 5;
    const int half = lane >> 4;   // 0: K pair {0,1}; 1: K pair {2,3}
    const int m    = lane & 15;   // A-matrix row (support within wave tile)

    float* __restrict__ wout = ws;                 // w array at ws[0]
    float partAcc = 0.0f;                          // this lane's share of sum(w)

    for (int pass = 0; pass < PASSES; ++pass) {
        const int sBase = jt * TS + pass * (K1_WAVES * SUPW) + wave * SUPW;
        const float* __restrict__ Srow =
            sup + ((size_t)batch * NS + (size_t)(sBase + m)) * ND + 2 * half;

        v8f   acc = {};
        float sq  = 0.0f;   // this lane's half of ||s_m||^2
        #pragma unroll 8
        for (int k0 = 0; k0 < ND; k0 += 4) {
            // A (16x4 f32, ISA 7.12.2): lane half selects K pair, 2 floats/lane
            v2f a  = *(const v2f*)(Srow + k0);
            // B (4x16): broadcast q over all N columns; lane half selects K pair
            v2f bq = *(const v2f*)(&qlds[k0 + 2 * half]);
            sq = fmaf(a.x, a.x, sq);
            sq = fmaf(a.y, a.y, sq);
            // D = A*B + C  -> every column of D holds dot(q, s_m)
            acc = __builtin_amdgcn_wmma_f32_16x16x4_f32(
                false, a, false, bq, (short)0, acc, false, false);
        }

        // combine the two K-halves of ||s_m||^2 (lanes L and L^16 share m)
        const float sq_full = sq + __shfl_xor(sq, 16, 32);
        // lane L needs norms of supports (8*half + r); held by lane (8*half + r)
        float sn[8];
        #pragma unroll
        for (int r = 0; r < 8; ++r) sn[r] = __shfl(sq_full, 8 * half + r, 32);

        if (m == 0) {   // lanes 0 and 16: D column N==0 carries the dots
            const int jb = sBase + 8 * half;   // acc[r] is M = r + 8*half
            #pragma unroll
            for (int r = 0; r < 8; ++r) {
                float d2 = qn + sn[r] - 2.0f * acc[r];
                d2 = fmaxf(d2, 0.0f);
                const float wv = expf(-sqrtf(d2));
                wout[(size_t)batch * NS + jb + r] = wv;
                partAcc += wv;
            }
        }
    }

    // deterministic per-block sum of w (fixed order, no atomics)
    if (m == 0) wpart[wave * 2 + half] = partAcc;
    __syncthreads();
    if (tid == 0) {
        float s = 0.0f;
        for (int i = 0; i < 2 * K1_WAVES; ++i) s += wpart[i];
        ws[WS_SUMS + batch * JT + jt] = s;
    }
}

// ---------------------------------------------------------------------------
// Kernel 2: partial out[c] = sum_j w_j * labels[j][c] over one support tile.
// Fully coalesced streaming of the 200 MB labels array.
// ---------------------------------------------------------------------------
__global__ __launch_bounds__(128)
void nw_combine_kernel(const float* __restrict__ labels,
                       const float* __restrict__ w,
                       float* __restrict__ pout) {
    __shared__ float wl[TS];
    const int tid   = threadIdx.x;
    const int jt    = blockIdx.x;
    const int batch = blockIdx.y;

    const float* __restrict__ wsrc = w + (size_t)batch * NS + (size_t)jt * TS;
    for (int i = tid; i < TS; i += 128) wl[i] = wsrc[i];
    __syncthreads();

    if (tid < NC) {
        const float* __restrict__ lab =
            labels + ((size_t)batch * NS + (size_t)jt * TS) * NC + tid;
        float a0 = 0.f, a1 = 0.f, a2 = 0.f, a3 = 0.f;
        for (int j = 0; j < TS; j += 4) {
            a0 = fmaf(wl[j + 0], lab[(size_t)(j + 0) * NC], a0);
            a1 = fmaf(wl[j + 1], lab[(size_t)(j + 1) * NC], a1);
            a2 = fmaf(wl[j + 2], lab[(size_t)(j + 2) * NC], a2);
            a3 = fmaf(wl[j + 3], lab[(size_t)(j + 3) * NC], a3);
        }
        pout[(size_t)(batch * JT + jt) * 128 + tid] = (a0 + a1) + (a2 + a3);
    }
}

// ---------------------------------------------------------------------------
// Kernel 3: out[b][c] = sum_tiles pout / sum_tiles sums
// ---------------------------------------------------------------------------
__global__ __launch_bounds__(128)
void nw_finalize_kernel(const float* __restrict__ ws, float* __restrict__ out) {
    const int batch = blockIdx.x;
    const int c     = threadIdx.x;
    float sumw = 0.0f;
    for (int t = 0; t < JT; ++t) sumw += ws[WS_SUMS + batch * JT + t];
    if (c < NC) {
        float v = 0.0f;
        for (int t = 0; t < JT; ++t)
            v += ws[WS_POUT + (size_t)(batch * JT + t) * 128 + c];
        out[batch * NC + c] = v / sumw;
    }
}

extern "C" void kernel_launch(void* const* d_in, const int* in_sizes, int n_in,
                              void* d_out, int out_size, void* d_ws, size_t ws_size,
                              hipStream_t stream) {
    const float* q      = (const float*)d_in[0];   // (64, 256)
    const float* sup    = (const float*)d_in[1];   // (64, 8192, 256)
    const float* labels = (const float*)d_in[2];   // (64, 8192, 100)
    float*       out    = (float*)d_out;           // (64, 100)
    float*       ws     = (float*)d_ws;            // >= ~2.4 MB used

    dim3 g1(JT, NB);
    nw_dist_kernel<<<g1, K1_THREADS, 0, stream>>>(q, sup, ws);

    dim3 g2(JT, NB);
    nw_combine_kernel<<<g2, 128, 0, stream>>>(labels, ws, ws + WS_POUT);

    nw_finalize_kernel<<<NB, 128, 0, stream>>>(ws, out);
}